// RelationRepresentation_22986664968864
// MI455X (gfx1250) — compile-verified
//
#include <hip/hip_runtime.h>

// ---------------------------------------------------------------------------
// RelationRepresentation for MI455X (gfx1250, wave32).
//   B=32, L=4096, D=1024.
//   Stage 1: three masked row-means of h -> combined (32 x 3072)   [HBM bound]
//   Stage 2: relu(combined @ W1 + b1)   -> hmid     (32 x 1024)   [f32 WMMA]
//   Stage 3: hmid @ W2 + b2             -> out      (32 x 1024)   [f32 WMMA]
// ---------------------------------------------------------------------------

typedef float v2f __attribute__((ext_vector_type(2)));
typedef float v8f __attribute__((ext_vector_type(8)));

#define B_DIM 32
#define L_DIM 4096
#define D_DIM 1024

// ---------------------------------------------------------------------------
// Kernel 1: masked means. grid = (B, 3), block = 256 threads.
// seg 0 = span1 [s1, e1+1), seg 1 = context [e1+1, s2), seg 2 = span2 [s2, e2+1)
// Each thread owns 4 consecutive columns (float4, 16B aligned, coalesced rows).
// ---------------------------------------------------------------------------
__global__ void span_mean_kernel(const float* __restrict__ h,
                                 const long long* __restrict__ idx1,
                                 const long long* __restrict__ idx2,
                                 float* __restrict__ combined) {
  const int b   = blockIdx.x;
  const int seg = blockIdx.y;

  const long long s1 = idx1[2 * b];
  const long long e1 = idx1[2 * b + 1];
  const long long s2 = idx2[2 * b];
  const long long e2 = idx2[2 * b + 1];

  long long lo, hi;
  if (seg == 0)      { lo = s1;     hi = e1 + 1; }
  else if (seg == 1) { lo = e1 + 1; hi = s2;     }
  else               { lo = s2;     hi = e2 + 1; }
  if (lo < 0) lo = 0;
  if (hi > L_DIM) hi = L_DIM;

  long long cnt_i = (hi > lo) ? (hi - lo) : 0;
  float inv = 1.0f / fmaxf((float)cnt_i, 1.0f);

  const int col4 = threadIdx.x * 4;
  const float* base = h + (size_t)b * L_DIM * D_DIM + col4;

  float4 acc = make_float4(0.f, 0.f, 0.f, 0.f);
  for (long long r = lo; r < hi; ++r) {
    const float4 v = *(const float4*)(base + (size_t)r * D_DIM);
    acc.x += v.x; acc.y += v.y; acc.z += v.z; acc.w += v.w;
  }

  float* out = combined + (size_t)b * 3072 + (size_t)seg * 1024 + col4;
  out[0] = acc.x * inv;
  out[1] = acc.y * inv;
  out[2] = acc.z * inv;
  out[3] = acc.w * inv;
}

// ---------------------------------------------------------------------------
// Kernels 2/3: 32xK @ KxN GEMM (N = 1024) with V_WMMA_F32_16X16X4_F32.
// One wave per 16x16 output tile: 2 M-tiles x 64 N-tiles = 128 waves.
// Launch: <<<32, 128>>> (4 waves per block). No divergence -> EXEC all ones.
//
// A-operand layout (16x4 f32, ISA 7.12.2): lanes 0-15 hold M=lane, K=k0/k0+1;
// lanes 16-31 hold M=lane-16, K=k0+2/k0+3.  B mirrors with N in place of M.
// C/D layout: VGPR r holds M = r + 8*(lane>=16), N = lane&15 (+tile offsets).
// ---------------------------------------------------------------------------
__global__ void wmma_gemm32_kernel(const float* __restrict__ A,   // 32 x K row-major
                                   const float* __restrict__ W,   // K x 1024 row-major
                                   const float* __restrict__ bias,// 1024
                                   float* __restrict__ out,       // 32 x 1024
                                   int K, int apply_relu) {
  const int lane    = threadIdx.x & 31;
  const int wave    = threadIdx.x >> 5;
  const int tile    = blockIdx.x * 4 + wave;  // 0..127
  const int tileM   = tile >> 6;              // 0..1
  const int tileN   = tile & 63;              // 0..63
  const int halfsel = lane >> 4;              // 0 | 1
  const int l       = lane & 15;
  const int m       = tileM * 16 + l;         // A row for this lane
  const int n       = tileN * 16 + l;         // B / out column for this lane
  const int koff    = halfsel * 2;            // K sub-offset per lane half

  const float* __restrict__ arow = A + (size_t)m * K;

  v8f c = {};
#pragma unroll 4
  for (int k0 = 0; k0 < K; k0 += 4) {
    const int k = k0 + koff;
    v2f a;
    a.x = arow[k];
    a.y = arow[k + 1];
    v2f bm;
    bm.x = W[(size_t)k * 1024 + n];
    bm.y = W[(size_t)(k + 1) * 1024 + n];
    // D = A(16x4) * B(4x16) + C ; emits v_wmma_f32_16x16x4_f32
    c = __builtin_amdgcn_wmma_f32_16x16x4_f32(
        /*neg_a=*/false, a, /*neg_b=*/false, bm,
        /*c_mod=*/(short)0, c, /*reuse_a=*/false, /*reuse_b=*/false);
  }

  const float bv = bias[n];
#pragma unroll
  for (int r = 0; r < 8; ++r) {
    const int row = tileM * 16 + r + halfsel * 8;
    float v = c[r] + bv;
    if (apply_relu) v = fmaxf(v, 0.0f);
    out[(size_t)row * 1024 + n] = v;
  }
}

// ---------------------------------------------------------------------------
// Host launcher
// ---------------------------------------------------------------------------
extern "C" void kernel_launch(void* const* d_in, const int* in_sizes, int n_in,
                              void* d_out, int out_size, void* d_ws, size_t ws_size,
                              hipStream_t stream) {
  const float*     h    = (const float*)d_in[0];      // (32, 4096, 1024) f32
  const long long* idx1 = (const long long*)d_in[1];  // (32, 1, 2) i64
  const long long* idx2 = (const long long*)d_in[2];  // (32, 1, 2) i64
  const float*     W1   = (const float*)d_in[3];      // (3072, 1024) f32
  const float*     b1   = (const float*)d_in[4];      // (1024,) f32
  const float*     W2   = (const float*)d_in[5];      // (1024, 1024) f32
  const float*     b2   = (const float*)d_in[6];      // (1024,) f32
  float*           out  = (float*)d_out;              // (32, 1, 1024) f32

  float* combined = (float*)d_ws;                     // 32*3072 f32 (384 KB)
  float* hmid     = combined + (size_t)B_DIM * 3072;  // 32*1024 f32 (128 KB)

  // Stage 1: masked means into combined
  span_mean_kernel<<<dim3(B_DIM, 3), 256, 0, stream>>>(h, idx1, idx2, combined);

  // Stage 2: hmid = relu(combined @ W1 + b1)
  wmma_gemm32_kernel<<<32, 128, 0, stream>>>(combined, W1, b1, hmid, 3072, 1);

  // Stage 3: out = hmid @ W2 + b2
  wmma_gemm32_kernel<<<32, 128, 0, stream>>>(hmid, W2, b2, out, 1024, 0);
}